// MFA_77910706750166
// MI455X (gfx1250) — compile-verified
//
#include <hip/hip_runtime.h>
#include <math.h>

typedef __attribute__((ext_vector_type(2))) float v2f;
typedef __attribute__((ext_vector_type(4))) float v4f;
typedef __attribute__((ext_vector_type(8))) float v8f;

#define N_PTS  131072
#define KCOMP  64
#define DF     512
#define LF     10
#define TN     32              // points per block

#define W1_ELEMS (DF * 1024)   // 524288 floats (2 MB)
#define W2_ELEMS (DF * 64)     // 32768 floats (128 KB)
#define PAR_STRIDE 128         // per-k: [0..99]=iL, [100..109]=c, [110]=CONST

// ---------------------------------------------------------------------------
// Kernel 1: per-component precompute. One block per component k.
// W1 layout (K-pair interleaved so a-fragments load as even-aligned pairs):
//   elem(dd, k, row) -> (dd>>1)*2048 + ((k>>4)*256 + row*16 + (k&15))*2 + (dd&1)
// W2 layout: elem(dd, k) -> (dd>>1)*128 + k*2 + (dd&1)
// ---------------------------------------------------------------------------
__global__ __launch_bounds__(256) void mfa_prep(const float* __restrict__ MU,
                                                const float* __restrict__ A,
                                                const float* __restrict__ Dm,
                                                const float* __restrict__ PI,
                                                float* __restrict__ ws)
{
    const int k   = blockIdx.x;
    const int tid = threadIdx.x;
    float* W1  = ws;
    float* W2  = ws + W1_ELEMS;
    float* PAR = ws + W1_ELEMS + W2_ELEMS;

    // 0..54: Gram upper-tri, 55..64: c, 65: cq, 66: sum log iD
    __shared__ float sred[67];
    if (tid < 67) sred[tid] = 0.0f;
    __syncthreads();

    float g[55];
    float cv[LF];
    #pragma unroll
    for (int i = 0; i < 55; ++i) g[i] = 0.0f;
    #pragma unroll
    for (int i = 0; i < LF; ++i) cv[i] = 0.0f;
    float cq = 0.0f, slog = 0.0f;

    const int kslot = ((k >> 4) * 256 + (k & 15)) * 2;
    for (int dd = tid; dd < DF; dd += 256) {
        float Dv = Dm[k * DF + dd];
        float iD = 1.0f / (Dv * Dv);
        float mu = MU[k * DF + dd];
        float a[LF];
        #pragma unroll
        for (int l = 0; l < LF; ++l) a[l] = A[((size_t)k * DF + dd) * LF + l];

        float* w1 = W1 + (size_t)(dd >> 1) * 2048 + kslot + (dd & 1);
        #pragma unroll
        for (int r = 0; r < LF; ++r) w1[r * 32] = iD * a[r];   // B rows
        w1[10 * 32] = iD * mu;                                  // iD*MU (s2)
        #pragma unroll
        for (int r = 11; r < 16; ++r) w1[r * 32] = 0.0f;        // pad rows
        W2[(dd >> 1) * 128 + k * 2 + (dd & 1)] = iD;            // s1 weights

        int idx = 0;
        #pragma unroll
        for (int l = 0; l < LF; ++l) {
            float ail = a[l] * iD;
            #pragma unroll
            for (int m = l; m < LF; ++m) g[idx++] += ail * a[m];
            cv[l] += ail * mu;                                  // c = B^T MU
        }
        cq   += iD * mu * mu;
        slog += logf(iD);
    }
    #pragma unroll
    for (int i = 0; i < 55; ++i) atomicAdd(&sred[i], g[i]);
    #pragma unroll
    for (int i = 0; i < LF; ++i) atomicAdd(&sred[55 + i], cv[i]);
    atomicAdd(&sred[65], cq);
    atomicAdd(&sred[66], slog);
    __syncthreads();

    if (tid == 0) {
        float Lm[LF][LF], inv[LF][LF];
        int idx = 0;
        for (int l = 0; l < LF; ++l)
            for (int m = l; m < LF; ++m) {
                float v = sred[idx++];
                Lm[l][m] = v; Lm[m][l] = v;
            }
        for (int i = 0; i < LF; ++i) {
            Lm[i][i] += 1.0f;
            for (int j = 0; j < LF; ++j) inv[i][j] = (i == j) ? 1.0f : 0.0f;
        }
        float logdet = 0.0f;                 // Gauss-Jordan on SPD
        for (int p = 0; p < LF; ++p) {
            float piv = Lm[p][p];
            logdet += logf(piv);
            float r = 1.0f / piv;
            for (int j = 0; j < LF; ++j) { Lm[p][j] *= r; inv[p][j] *= r; }
            for (int i = 0; i < LF; ++i) {
                if (i == p) continue;
                float f = Lm[i][p];
                for (int j = 0; j < LF; ++j) {
                    Lm[i][j]  -= f * Lm[p][j];
                    inv[i][j] -= f * inv[p][j];
                }
            }
        }
        float* pk = PAR + k * PAR_STRIDE;
        for (int i = 0; i < LF; ++i)
            for (int j = 0; j < LF; ++j) pk[i * LF + j] = inv[i][j];
        for (int l = 0; l < LF; ++l) pk[100 + l] = sred[55 + l];
        const float LOG2PI = 1.8378770664093453f;
        pk[110] = PI[k] - 0.5f * ((float)DF * LOG2PI + logdet - sred[66] + sred[65]);
    }
}

// ---------------------------------------------------------------------------
// Kernel 2: main WMMA kernel. One block per 32-point tile; 8 wave32.
// Wave w: component group (w&3)*16..+15, n-subtile (w>>2)*16..+15.
// ---------------------------------------------------------------------------
__device__ inline v8f wmma4(v2f a, v2f b, v8f c) {
    return __builtin_amdgcn_wmma_f32_16x16x4_f32(false, a, false, b,
                                                 (short)0, c, false, false);
}

__global__ __launch_bounds__(256) void mfa_main(const float* __restrict__ x,
                                                const float* __restrict__ ws,
                                                float* __restrict__ out)
{
    const float* W1  = ws;
    const float* W2  = ws + W1_ELEMS;
    const float* PAR = ws + W1_ELEMS + W2_ELEMS;

    // phase 1: xs[512][33] (16896 f); phase 2: ts[768][33] (25344 f) — unioned
    __shared__ float smem[25344];
    float* xs = smem;
    float* ts = smem;

    const int tid  = threadIdx.x;
    const int lane = tid & 31;
    const int wave = tid >> 5;
    const int nl   = lane & 15;               // M / N index within tile
    const int h2   = (lane < 16) ? 0 : 2;     // K-pair select for fragments
    const int h8   = (lane < 16) ? 0 : 8;     // C-tile row-half select
    const int wgrp = wave & 3;                // component group (16 comps)
    const int nsub = wave >> 2;               // n-subtile (0 or 1)
    const int n0   = blockIdx.x * TN;

    // ---- load 32 x-rows transposed into LDS (coalesced float4 reads) ----
    for (int it = 0; it < 16; ++it) {
        int idx = it * 256 + tid;             // 0..4095
        int n   = idx >> 7;                   // 0..31
        int q   = idx & 127;                  // float4 index within row
        float4 xv = *(const float4*)(x + (size_t)(n0 + n) * DF + q * 4);
        xs[(q * 4 + 0) * 33 + n] = xv.x;
        xs[(q * 4 + 1) * 33 + n] = xv.y;
        xs[(q * 4 + 2) * 33 + n] = xv.z;
        xs[(q * 4 + 3) * 33 + n] = xv.w;
    }
    __syncthreads();

    v8f acc[16] = {};    // 16 component tiles (rows 0..9=t, 10=s2)
    v8f acc1   = {};     // s1 tile: comps wgrp*16..+15, this n-subtile

    const int abase = wgrp * 512 + nl * 32;   // pair-interleaved W1 frag base
    const int a2off = (wgrp * 16 + nl) * 2;   // W2 fragment offset
    const int bcol  = nsub * 16 + nl;         // xs column for this wave

    #pragma unroll 1
    for (int kk = 0; kk < DF; kk += 4) {
        // B fragment (shared by all 17 WMMAs of this wave)
        v2f b;
        b.x = xs[(kk + h2    ) * 33 + bcol];
        b.y = xs[(kk + h2 + 1) * 33 + bcol];

        // A fragments: 32 contiguous floats = 16 even-aligned v2f pairs
        const v4f* ap = (const v4f*)(W1 + (size_t)(kk + h2) * 1024 + abase);
        #pragma unroll
        for (int c = 0; c < 8; ++c) {
            v4f q = ap[c];
            v2f a0 = __builtin_shufflevector(q, q, 0, 1);
            v2f a1 = __builtin_shufflevector(q, q, 2, 3);
            acc[2 * c    ] = wmma4(a0, b, acc[2 * c    ]);
            acc[2 * c + 1] = wmma4(a1, b, acc[2 * c + 1]);
        }

        // s1 = sum iD * x^2 : square b in registers, one dense tile per wave
        v2f bq; bq.x = b.x * b.x; bq.y = b.y * b.y;
        v2f a2 = *(const v2f*)(W2 + (size_t)(kk + h2) * 64 + a2off);
        acc1 = wmma4(a2, bq, acc1);
    }
    __syncthreads();   // xs dead; reuse smem as ts

    // ---- spill accumulator tiles to LDS (12 useful rows per component) ----
    #pragma unroll
    for (int cg = 0; cg < 16; ++cg) {
        int k = wgrp * 16 + cg;
        #pragma unroll
        for (int v = 0; v < 8; ++v) {
            int row = v + h8;
            if (row < 11) ts[(k * 12 + row) * 33 + nsub * 16 + nl] = acc[cg][v];
        }
    }
    #pragma unroll
    for (int v = 0; v < 8; ++v) {
        int kc = wgrp * 16 + v + h8;
        ts[(kc * 12 + 11) * 33 + nsub * 16 + nl] = acc1[v];
    }
    __syncthreads();

    // ---- epilogue: q2 = t^T iL t, combine, coalesced (N,K) store ----
    for (int j = 0; j < 8; ++j) {
        int idx = j * 256 + tid;              // 0..2047
        int k   = idx & 63;
        int n   = idx >> 6;                   // 0..31
        const float* pk = PAR + k * PAR_STRIDE;
        float t[LF];
        #pragma unroll
        for (int l = 0; l < LF; ++l)
            t[l] = ts[(k * 12 + l) * 33 + n] - pk[100 + l];
        float s2 = ts[(k * 12 + 10) * 33 + n];
        float s1 = ts[(k * 12 + 11) * 33 + n];
        float q2 = 0.0f;
        #pragma unroll
        for (int i = 0; i < LF; ++i) {
            float wi = 0.0f;
            #pragma unroll
            for (int jj = 0; jj < LF; ++jj) wi += pk[i * LF + jj] * t[jj];
            q2 += t[i] * wi;
        }
        out[(size_t)(n0 + n) * KCOMP + k] =
            pk[110] - 0.5f * (s1 - 2.0f * s2) + 0.5f * q2;
    }
}

// ---------------------------------------------------------------------------
extern "C" void kernel_launch(void* const* d_in, const int* in_sizes, int n_in,
                              void* d_out, int out_size, void* d_ws, size_t ws_size,
                              hipStream_t stream) {
    const float* x  = (const float*)d_in[0];
    const float* MU = (const float*)d_in[1];
    const float* A  = (const float*)d_in[2];
    const float* Dm = (const float*)d_in[3];
    const float* PI = (const float*)d_in[4];
    float* out = (float*)d_out;
    float* ws  = (float*)d_ws;   // needs ~2.2 MB

    hipLaunchKernelGGL(mfa_prep, dim3(KCOMP), dim3(256), 0, stream,
                       MU, A, Dm, PI, ws);
    hipLaunchKernelGGL(mfa_main, dim3(N_PTS / TN), dim3(256), 0, stream,
                       x, ws, out);
}